// BinomialResample_47665547051439
// MI455X (gfx1250) — compile-verified
//
#include <hip/hip_runtime.h>
#include <hip/hip_bf16.h>
#include <cstdint>

#define BLOCK 256
#define F4_PER_TILE 256   // one float4 per thread per tile -> 4KB tile

typedef __attribute__((ext_vector_type(4))) float v4f;  // native vector for NT store

// ---------------- Philox 4x32-10 (stateless, counter-based) ----------------
__device__ __forceinline__ void philox4x32(uint32_t c0, uint32_t c1, uint32_t c2, uint32_t c3,
                                           uint32_t k0, uint32_t k1, uint32_t out[4]) {
#pragma unroll
    for (int r = 0; r < 10; ++r) {
        uint32_t lo0 = 0xD2511F53u * c0;
        uint32_t hi0 = __umulhi(0xD2511F53u, c0);
        uint32_t lo1 = 0xCD9E8D57u * c2;
        uint32_t hi1 = __umulhi(0xCD9E8D57u, c2);
        uint32_t n0 = hi1 ^ c1 ^ k0;
        uint32_t n1 = lo1;
        uint32_t n2 = hi0 ^ c3 ^ k1;
        uint32_t n3 = lo0;
        c0 = n0; c1 = n1; c2 = n2; c3 = n3;
        k0 += 0x9E3779B9u;
        k1 += 0xBB67AE85u;
    }
    out[0] = c0; out[1] = c1; out[2] = c2; out[3] = c3;
}

__device__ __forceinline__ float u01(uint32_t x) {
    // uniform in [0, 1) with 24-bit mantissa
    return (float)(x >> 8) * (1.0f / 16777216.0f);
}

// ---------------- Binomial sampler ----------------
// p in [0.5, 0.99]; use Binomial(n,p) = n - Binomial(n, q), q = 1-p <= 0.5,
// and invert the CDF of Binomial(n, q) starting at k=0 (expected iters ~ n*q).
__device__ __forceinline__ float binom_sample(float n, float p, float u) {
    float q    = 1.0f - p;
    float odds = __fdividef(q, p);                 // q / (1-q)
    float f    = __expf(n * __logf(p));            // pmf(0) = (1-q)^n = p^n
    float c    = f;
    float k    = 0.0f;
    while (u > c && k < n) {
        k += 1.0f;
        f *= __fdividef((n - k + 1.0f) * odds, k); // pmf recurrence
        c += f;
    }
    return n - k;
}

// ---------------- Main kernel: async LDS double-buffered stream ----------------
__global__ void __launch_bounds__(BLOCK)
binomial_resample_kernel(const float* __restrict__ x, float* __restrict__ y, int nTiles) {
    __shared__ float4 lbuf[2][F4_PER_TILE];

    const int tid = threadIdx.x;
    int tile = blockIdx.x;
    if (tile >= nTiles) return;

    const float4* __restrict__ x4 = (const float4*)x;
    v4f* __restrict__ y4 = (v4f*)y;

    auto issue_async = [&](int t, int par) {
        // LDS byte offset of this lane's slot (flat shared addr low 32 bits == LDS offset)
        uint32_t lds_off = (uint32_t)(uintptr_t)(&lbuf[par][tid]);
        const float4* src = x4 + ((long long)t * F4_PER_TILE + tid);
        asm volatile("global_load_async_to_lds_b128 %0, %1, off"
                     :: "v"(lds_off), "v"(src)
                     : "memory");
    };

    issue_async(tile, 0);
    int par = 0;

    while (true) {
        int next = tile + (int)gridDim.x;
        if (next < nTiles) {
            issue_async(next, par ^ 1);
            asm volatile("s_wait_asynccnt 1" ::: "memory"); // tile `tile` has landed
        } else {
            asm volatile("s_wait_asynccnt 0" ::: "memory");
        }

        const float4 v = lbuf[par][tid];

        const uint32_t idx = (uint32_t)tile * F4_PER_TILE + (uint32_t)tid; // < 2^26, unique
        uint32_t rp[4], ru[4];
        philox4x32(idx, 0x13371337u, 0xDEADBEEFu, 0x0BADF00Du,
                   0xA5A5A5A5u, 0x3C6EF372u, rp);          // -> p draws
        philox4x32(idx, 0x74545454u, 0x9E3779B9u, 0x85EBCA6Bu,
                   0xC2B2AE35u, 0x27220A95u, ru);          // -> inversion uniforms

        v4f out;
        out.x = binom_sample(v.x, 0.5f + 0.49f * u01(rp[0]), u01(ru[0]));
        out.y = binom_sample(v.y, 0.5f + 0.49f * u01(rp[1]), u01(ru[1]));
        out.z = binom_sample(v.z, 0.5f + 0.49f * u01(rp[2]), u01(ru[2]));
        out.w = binom_sample(v.w, 0.5f + 0.49f * u01(rp[3]), u01(ru[3]));

        // Streaming output: non-temporal store (TH=NT) — no reuse, don't pollute L2.
        __builtin_nontemporal_store(out, &y4[(long long)tile * F4_PER_TILE + tid]);

        if (next >= nTiles) break;
        tile = next;
        par ^= 1;
    }
}

// ---------------- Scalar tail (defensive; N is exactly divisible in practice) ----------------
__global__ void __launch_bounds__(BLOCK)
binomial_resample_tail_kernel(const float* __restrict__ x, float* __restrict__ y,
                              int start, int n) {
    int i = start + blockIdx.x * BLOCK + threadIdx.x;
    if (i >= n) return;
    uint32_t r[4];
    philox4x32((uint32_t)i, 0x7A117A11u, 0xDEADBEEFu, 0x0BADF00Du,
               0xA5A5A5A5u, 0x3C6EF372u, r);
    float p = 0.5f + 0.49f * u01(r[0]);
    y[i] = binom_sample(x[i], p, u01(r[1]));
}

extern "C" void kernel_launch(void* const* d_in, const int* in_sizes, int n_in,
                              void* d_out, int out_size, void* d_ws, size_t ws_size,
                              hipStream_t stream) {
    const float* x = (const float*)d_in[0];
    float* y = (float*)d_out;
    const int n = in_sizes[0];                 // 4096*36601 = 149,917,696

    const int elemsPerTile = 4 * F4_PER_TILE;  // 1024
    const int nTiles = n / elemsPerTile;       // exact: 146,404
    const int tail = nTiles * elemsPerTile;

    if (nTiles > 0) {
        int grid = nTiles < 4096 ? nTiles : 4096;
        binomial_resample_kernel<<<dim3(grid), dim3(BLOCK), 0, stream>>>(x, y, nTiles);
    }
    if (tail < n) {
        int rem = n - tail;
        binomial_resample_tail_kernel<<<dim3((rem + BLOCK - 1) / BLOCK), dim3(BLOCK), 0, stream>>>(
            x, y, tail, n);
    }
}